// DistanceTransformLoss_18691697672891
// MI455X (gfx1250) — compile-verified
//
#include <hip/hip_runtime.h>
#include <hip/hip_bf16.h>
#include <stdint.h>

// DistanceTransformLoss for MI455X (gfx1250).
//  - Kernel 1: fused CE(sum,ignore=255) + argmax + target-border + dist init (one 90MB pass)
//  - Kernel 2: chessboard DT: 48 launches x 8 min-pool steps in LDS (64x64 tile + 8 halo),
//              tile loaded AND stored via Tensor Data Mover (tensor_load_to_lds /
//              tensor_store_from_lds), global ping-pong between two 777-padded arrays.
//  - Kernel 3: pred-border * dist partial sums
//  - Kernel 4: deterministic final reduction -> scalar

namespace {
constexpr int   B  = 8, C = 19, H = 384, W = 384;
constexpr int   HW = H * W;
constexpr int   NPIX = B * HW;                 // 1,179,648
constexpr float BOUND = 777.0f;                // B + 1 + H + W
constexpr float BORDER_FACTOR = 0.2f;
constexpr int   IGNORE_INDEX = 255;

constexpr int P  = 8;                          // global halo ring width
constexpr int S  = 8;                          // min-pool steps per launch
constexpr int TS = 64;                         // interior tile
constexpr int TW = TS + 2 * S;                 // 80: loaded tile incl. halo
constexpr int Hp = H + 2 * P, Wp = W + 2 * P;  // 400
constexpr int NTILE = H / TS;                  // 6
constexpr int NITER = (H + S - 1) / S;         // 48 launches * 8 steps = 384
constexpr int CE_BLOCKS = NPIX / 256;          // 4608
}

typedef unsigned int uv4 __attribute__((ext_vector_type(4)));
typedef int          iv4 __attribute__((ext_vector_type(4)));
typedef int          iv8 __attribute__((ext_vector_type(8)));

#if __has_builtin(__builtin_amdgcn_tensor_load_to_lds) && \
    __has_builtin(__builtin_amdgcn_tensor_store_from_lds)
#define USE_TDM 1
#else
#define USE_TDM 0
#endif

#if USE_TDM
// Build D# groups for a 2D f32 tile of (tile_w x tile_h) within a row-major
// tensor of width Wp (elements), height B*Hp, stride Wp.
__device__ __forceinline__ void make_tdm_desc(uint64_t gaddr, unsigned lds,
                                              int tile_w, int tile_h,
                                              uv4& g0, iv8& g1) {
  // group0: [1:0]=count=1, [63:32]=lds_addr, [120:64]=global_addr, [127:126]=type=2
  g0[0] = 1u;
  g0[1] = lds;
  g0[2] = (unsigned int)(gaddr & 0xffffffffull);
  g0[3] = (unsigned int)((gaddr >> 32) & 0x01ffffffull) | (2u << 30);
  // group1: data_size=4B(2), tensor_dim0=Wp, tensor_dim1=B*Hp,
  //         tile_dim0, tile_dim1, tensor_dim0_stride=Wp
  const unsigned d0 = (unsigned)Wp, d1 = (unsigned)(B * Hp);
  g1[0] = (int)(2u << 16);
  g1[1] = (int)((d0 & 0xffffu) << 16);
  g1[2] = (int)(((d0 >> 16) & 0xffffu) | ((d1 & 0xffffu) << 16));
  g1[3] = (int)(((d1 >> 16) & 0xffffu) | ((unsigned)tile_w << 16));
  g1[4] = (int)((unsigned)tile_h);
  g1[5] = (int)((unsigned)Wp);      // dim0 stride (elements), low 32
  g1[6] = 0;
  g1[7] = 0;
}
#endif

// ---------------------------------------------------------------- fill 777
__global__ __launch_bounds__(256) void fill_kernel(float* __restrict__ p, int n, float v) {
  int i = blockIdx.x * 256 + threadIdx.x;
  if (i < n) p[i] = v;
}

// ------------------------------------------- CE + argmax + tgt border + init
__global__ __launch_bounds__(256) void ce_pred_init_kernel(
    const float* __restrict__ slices, const int* __restrict__ targets,
    float* __restrict__ distA, unsigned char* __restrict__ pred,
    float* __restrict__ ce_partials) {
  const int tid = blockIdx.x * 256 + threadIdx.x;
  const int b  = tid / HW;
  const int hw = tid - b * HW;
  const int h  = hw / W;
  const int w  = hw - h * W;

  const float* x = slices + (size_t)b * C * HW + hw;
  const int tgt = targets[tid];

  float vals[C];
  vals[0] = x[0];
  float m = vals[0];
  int   bi = 0;
  float xt = (tgt == 0) ? vals[0] : 0.0f;
#pragma unroll
  for (int c = 1; c < C; ++c) {
    float v = x[(size_t)c * HW];
    vals[c] = v;
    if (v > m) { m = v; bi = c; }
    if (c == tgt) xt = v;
  }
  float s = 0.0f;
#pragma unroll
  for (int c = 0; c < C; ++c) s += __expf(vals[c] - m);
  const bool valid = (tgt != IGNORE_INDEX);
  const float nll = valid ? (m + __logf(s) - xt) : 0.0f;

  pred[tid] = (unsigned char)bi;

  // target border: (forward-diff down + forward-diff right) != 0, zero-padded at ends
  const int tb = (h < H - 1) ? (targets[tid + W] - tgt) : 0;
  const int lr = (w < W - 1) ? (targets[tid + 1] - tgt) : 0;
  const bool isb = (tb + lr) != 0;
  distA[((size_t)b * Hp + (P + h)) * Wp + (P + w)] = isb ? 0.0f : BOUND;

  // deterministic block reduction of nll
  __shared__ float red[256];
  red[threadIdx.x] = nll;
  __syncthreads();
  for (int o = 128; o > 0; o >>= 1) {
    if (threadIdx.x < o) red[threadIdx.x] += red[threadIdx.x + o];
    __syncthreads();
  }
  if (threadIdx.x == 0) ce_partials[blockIdx.x] = red[0];
}

// ---------------------------- 8 chessboard min-pool steps on a 64x64 tile
__global__ __launch_bounds__(256) void dist_step_kernel(
    const float* __restrict__ src, float* __restrict__ dst) {
  __shared__ float buf[2][TW * TW];   // 2 * 25.6KB

  const int tx0 = blockIdx.x * TS;    // load-origin in padded coords
  const int ty0 = blockIdx.y * TS;
  const int b   = blockIdx.z;

  const uint64_t gaddr = (uint64_t)(uintptr_t)src +
      4ull * (((uint64_t)b * Hp + (uint64_t)ty0) * Wp + (uint64_t)tx0);

#if USE_TDM
  // Tensor Data Mover: 2D tile TWxTW (f32) from padded global array -> LDS.
  if ((threadIdx.x >> 5) == 0) {      // one wave issues the TDM op
    uv4 g0; iv8 g1;
    iv4 g2 = {0, 0, 0, 0}, g3 = {0, 0, 0, 0};
    iv8 g4 = {0, 0, 0, 0, 0, 0, 0, 0};
    make_tdm_desc(gaddr, (unsigned int)(uintptr_t)(&buf[0][0]), TW, TW, g0, g1);
    __builtin_amdgcn_tensor_load_to_lds(g0, g1, g2, g3, g4, 0);
    __builtin_amdgcn_s_wait_tensorcnt(0);
  }
#else
  for (int i = threadIdx.x; i < TW * TW; i += 256) {
    int r = i / TW, c = i - r * TW;
    buf[0][i] = src[((size_t)b * Hp + ty0 + r) * Wp + tx0 + c];
  }
#endif
  __syncthreads();

  int cur = 0;
#pragma unroll 1
  for (int step = 0; step < S; ++step) {
    const float* __restrict__ o = buf[cur];
    float* __restrict__ n = buf[cur ^ 1];
    for (int i = threadIdx.x; i < TW * TW; i += 256) {
      const int r = i / TW, c = i - r * TW;
      const float v = o[i];
      float nv = v;
      if (r > 0 && r < TW - 1 && c > 0 && c < TW - 1) {
        const float* up = o + i - TW;
        const float* dn = o + i + TW;
        float mr = fminf(fminf(up[-1], up[0]), up[1]);
        float mm = fminf(o[i - 1], o[i + 1]);
        float md = fminf(fminf(dn[-1], dn[0]), dn[1]);
        float m9 = fminf(fminf(mr, md), fminf(mm, v));
        nv = fminf(v, m9 + 1.0f);
      }
      n[i] = nv;
    }
    __syncthreads();
    cur ^= 1;
  }

  // write back interior TSxTS
  const float* __restrict__ fin = buf[cur];
#if USE_TDM
  // Compact interior into the spare buffer (TDM needs a linear LDS tile),
  // then DMA it out with tensor_store_from_lds.
  float* __restrict__ cbuf = buf[cur ^ 1];
  for (int i = threadIdx.x; i < TS * TS; i += 256) {
    const int r = i / TS, c = i - r * TS;
    cbuf[i] = fin[(P + r) * TW + (P + c)];
  }
  __syncthreads();
  if ((threadIdx.x >> 5) == 0) {
    const uint64_t daddr = (uint64_t)(uintptr_t)dst +
        4ull * (((uint64_t)b * Hp + (uint64_t)(ty0 + P)) * Wp + (uint64_t)(tx0 + P));
    uv4 g0; iv8 g1;
    iv4 g2 = {0, 0, 0, 0}, g3 = {0, 0, 0, 0};
    iv8 g4 = {0, 0, 0, 0, 0, 0, 0, 0};
    make_tdm_desc(daddr, (unsigned int)(uintptr_t)cbuf, TS, TS, g0, g1);
    __builtin_amdgcn_tensor_store_from_lds(g0, g1, g2, g3, g4, 0);
    __builtin_amdgcn_s_wait_tensorcnt(0);
  }
#else
  for (int i = threadIdx.x; i < TS * TS; i += 256) {
    const int r = i / TS, c = i - r * TS;
    dst[((size_t)b * Hp + ty0 + P + r) * Wp + (tx0 + P + c)] =
        fin[(P + r) * TW + (P + c)];
  }
#endif
}

// ----------------------------------------- pred border * dist, partial sums
__global__ __launch_bounds__(256) void border_loss_kernel(
    const unsigned char* __restrict__ pred, const float* __restrict__ dist,
    float* __restrict__ partials) {
  const int tid = blockIdx.x * 256 + threadIdx.x;
  const int b  = tid / HW;
  const int hw = tid - b * HW;
  const int h  = hw / W;
  const int w  = hw - h * W;

  const int pc = pred[tid];
  const int tb = (h < H - 1) ? ((int)pred[tid + W] - pc) : 0;
  const int lr = (w < W - 1) ? ((int)pred[tid + 1] - pc) : 0;
  const float d = ((tb + lr) != 0)
      ? dist[((size_t)b * Hp + (P + h)) * Wp + (P + w)] : 0.0f;

  __shared__ float red[256];
  red[threadIdx.x] = d;
  __syncthreads();
  for (int o = 128; o > 0; o >>= 1) {
    if (threadIdx.x < o) red[threadIdx.x] += red[threadIdx.x + o];
    __syncthreads();
  }
  if (threadIdx.x == 0) partials[blockIdx.x] = red[0];
}

// --------------------------------------------------- deterministic finalize
__global__ __launch_bounds__(256) void finalize_kernel(
    const float* __restrict__ cep, const float* __restrict__ bdp, int n,
    float* __restrict__ out) {
  __shared__ float red[256];
  float a = 0.0f;
  for (int i = threadIdx.x; i < n; i += 256)
    a += cep[i] + BORDER_FACTOR * bdp[i];
  red[threadIdx.x] = a;
  __syncthreads();
  for (int o = 128; o > 0; o >>= 1) {
    if (threadIdx.x < o) red[threadIdx.x] += red[threadIdx.x + o];
    __syncthreads();
  }
  if (threadIdx.x == 0) out[0] = red[0];
}

extern "C" void kernel_launch(void* const* d_in, const int* in_sizes, int n_in,
                              void* d_out, int out_size, void* d_ws, size_t ws_size,
                              hipStream_t stream) {
  const float* slices  = (const float*)d_in[0];
  const int*   targets = (const int*)d_in[1];

  char* ws = (char*)d_ws;
  size_t off = 0;
  float* distA = (float*)(ws + off); off += (size_t)B * Hp * Wp * sizeof(float);
  float* distB = (float*)(ws + off); off += (size_t)B * Hp * Wp * sizeof(float);
  unsigned char* pred = (unsigned char*)(ws + off); off += (size_t)NPIX;
  off = (off + 255) & ~(size_t)255;
  float* cep = (float*)(ws + off); off += (size_t)CE_BLOCKS * sizeof(float);
  float* bdp = (float*)(ws + off); off += (size_t)CE_BLOCKS * sizeof(float);

  // distA and distB are contiguous: one fill covers both (rings stay 777).
  const int ninit = 2 * B * Hp * Wp;
  fill_kernel<<<(ninit + 255) / 256, 256, 0, stream>>>(distA, ninit, BOUND);

  ce_pred_init_kernel<<<CE_BLOCKS, 256, 0, stream>>>(slices, targets, distA, pred, cep);

  float* bufs[2] = {distA, distB};
  for (int it = 0; it < NITER; ++it) {
    dist_step_kernel<<<dim3(NTILE, NTILE, B), 256, 0, stream>>>(
        bufs[it & 1], bufs[(it + 1) & 1]);
  }
  // NITER = 48 (even) -> final field back in distA
  border_loss_kernel<<<CE_BLOCKS, 256, 0, stream>>>(pred, bufs[NITER & 1], bdp);

  finalize_kernel<<<1, 256, 0, stream>>>(cep, bdp, CE_BLOCKS, (float*)d_out);
}